// DGCF_Encoder_12841952215156
// MI455X (gfx1250) — compile-verified
//
#include <hip/hip_runtime.h>
#include <math.h>

typedef float v2f __attribute__((ext_vector_type(2)));
typedef float v8f __attribute__((ext_vector_type(8)));

#define WG 256

__device__ __forceinline__ void atomic_fadd(float* p, float v) {
  // relaxed, device scope -> global_atomic_add_f32 on gfx1250
  __hip_atomic_fetch_add(p, v, __ATOMIC_RELAXED, __HIP_MEMORY_SCOPE_AGENT);
}

// Native CDNA5 transcendental tanh (TRANS32 op). The v_nop covers the
// documented trans-op result hazard (1 independent op / V_NOP before use);
// the compiler cannot schedule around an asm block's internal latency.
__device__ __forceinline__ float fast_tanh(float x) {
  float r;
  asm("v_tanh_f32 %0, %1\n\tv_nop\n\tv_nop" : "=v"(r) : "v"(x));
  return r;
}

// ---------------- setup ----------------
__global__ void k_init(const float* __restrict__ u, const float* __restrict__ it,
                       float* __restrict__ ego, float* __restrict__ acc,
                       int userElems, int total) {
  int i = blockIdx.x * blockDim.x + threadIdx.x;
  if (i >= total) return;
  float v = (i < userElems) ? u[i] : it[i - userElems];
  ego[i] = v;
  acc[i] = v;
}

__global__ void k_fill1(float* __restrict__ a, int n) {
  int i = blockIdx.x * blockDim.x + threadIdx.x;
  if (i < n) a[i] = 1.0f;
}

// ---------------- per-iteration stages ----------------
// softmax over factors of A[e], store normA into ew, scatter-add degrees
__global__ void k_softmax_dval(const float* __restrict__ A, const int* __restrict__ hl,
                               float* __restrict__ ew, float* __restrict__ dval, int E) {
  int e = blockIdx.x * blockDim.x + threadIdx.x;
  if (e >= E) return;
  float4 a = reinterpret_cast<const float4*>(A)[e];
  float m = fmaxf(fmaxf(a.x, a.y), fmaxf(a.z, a.w));
  float e0 = expf(a.x - m), e1 = expf(a.y - m), e2 = expf(a.z - m), e3 = expf(a.w - m);
  float inv = 1.0f / (e0 + e1 + e2 + e3);
  float4 nrm = make_float4(e0 * inv, e1 * inv, e2 * inv, e3 * inv);
  reinterpret_cast<float4*>(ew)[e] = nrm;
  float* d = dval + (size_t)hl[e] * 4;
  atomic_fadd(d + 0, nrm.x);
  atomic_fadd(d + 1, nrm.y);
  atomic_fadd(d + 2, nrm.z);
  atomic_fadd(d + 3, nrm.w);
}

// dinv = 1/sqrt(max(dval,1e-8)), in place
__global__ void k_dinv(float* __restrict__ d, int n) {
  int i = blockIdx.x * blockDim.x + threadIdx.x;
  if (i >= n) return;
  d[i] = 1.0f / sqrtf(fmaxf(d[i], 1e-8f));
}

// ew[e,f] = normA[e,f] * dinv[h,f] * dinv[t,f]   (in place on ew)
__global__ void k_edge_weight(float* __restrict__ ew, const float* __restrict__ dinv,
                              const int* __restrict__ hl, const int* __restrict__ tl, int E) {
  int e = blockIdx.x * blockDim.x + threadIdx.x;
  if (e >= E) return;
  float4 w = reinterpret_cast<float4*>(ew)[e];
  float4 dh = reinterpret_cast<const float4*>(dinv)[hl[e]];
  float4 dt = reinterpret_cast<const float4*>(dinv)[tl[e]];
  w.x *= dh.x * dt.x;
  w.y *= dh.y * dt.y;
  w.z *= dh.z * dt.z;
  w.w *= dh.w * dt.w;
  reinterpret_cast<float4*>(ew)[e] = w;
}

// factor_emb[h] += w[e,f] * ego[t] ; one thread per (edge, float4 chunk)
__global__ void k_message(const float* __restrict__ ew, const float* __restrict__ ego,
                          const int* __restrict__ hl, const int* __restrict__ tl,
                          float* __restrict__ fact, int E) {
  int tid = blockIdx.x * blockDim.x + threadIdx.x;
  int total = E * 16;
  if (tid >= total) return;
  int e = tid >> 4, q = tid & 15, f = q >> 2;
  int h = hl[e], t = tl[e];
  float w = ew[(size_t)e * 4 + f];
  float4 v = reinterpret_cast<const float4*>(ego)[(size_t)t * 16 + q];
  float* dst = fact + (size_t)h * 64 + q * 4;
  atomic_fadd(dst + 0, w * v.x);
  atomic_fadd(dst + 1, w * v.y);
  atomic_fadd(dst + 2, w * v.z);
  atomic_fadd(dst + 3, w * v.w);
}

// out[n,f] = 1 / max(||x[n, f*16 : f*16+16]||, 1e-12)
__global__ void k_norminv(const float* __restrict__ x, float* __restrict__ out, int NF) {
  int i = blockIdx.x * blockDim.x + threadIdx.x;
  if (i >= NF) return;
  int n = i >> 2, f = i & 3;
  const float4* p = reinterpret_cast<const float4*>(x + (size_t)n * 64 + (size_t)f * 16);
  float s = 0.f;
#pragma unroll
  for (int k = 0; k < 4; ++k) {
    float4 v = p[k];
    s += v.x * v.x + v.y * v.y + v.z * v.z + v.w * v.w;
  }
  out[i] = 1.0f / fmaxf(sqrtf(s), 1e-12f);
}

// Routing update via matrix core: A[e,f] += dot(head_hat, tanh(tail_hat)).
// One wave handles 16 edges; for each factor, 4 chained V_WMMA_F32_16X16X4_F32
// (K=4 f32) build the 16x16 Gram matrix; we keep the diagonal.
// A and B fragments use identical (lane-half, VGPR) -> K offset placement, so
// any consistent K permutation in the HW layout cancels on the diagonal.
__global__ void __launch_bounds__(WG) k_updateA(
    const float* __restrict__ fact, const float* __restrict__ ego,
    const float* __restrict__ hninv, const float* __restrict__ tninv,
    const int* __restrict__ hl, const int* __restrict__ tl,
    float* __restrict__ A, int E, int nGroups) {
  int tid = blockIdx.x * blockDim.x + threadIdx.x;
  int wave = tid >> 5;
  int lane = tid & 31;
  if (wave >= nGroups) return;  // wave-uniform exit, EXEC stays full for WMMA

  int r = lane & 15;                 // matrix row this lane feeds (M for A, N for B)
  int eIdx = wave * 16 + r;
  int e = (eIdx < E) ? eIdx : (E - 1);  // clamp (select, no divergence)
  int h = hl[e], t = tl[e];
  int koff = (lane >> 4) << 1;       // lanes 0-15 -> K{0,1}, lanes 16-31 -> K{2,3}
  const float* Hrow = fact + (size_t)h * 64;
  const float* Trow = ego + (size_t)t * 64;
  const float* hn4 = hninv + (size_t)h * 4;
  const float* tn4 = tninv + (size_t)t * 4;

  bool active = (lane < 8) || (lane >= 24);          // lanes holding diag elements
  int outRow = (lane < 8) ? lane : (lane - 16);      // which edge's diagonal
  int eOut = wave * 16 + outRow;

#pragma unroll
  for (int f = 0; f < 4; ++f) {
    float hn = hn4[f];
    float tn = tn4[f];
    v8f c = {0.f, 0.f, 0.f, 0.f, 0.f, 0.f, 0.f, 0.f};
#pragma unroll
    for (int s = 0; s < 4; ++s) {
      int o = f * 16 + s * 4 + koff;
      v2f av, bv;
      av.x = Hrow[o] * hn;
      av.y = Hrow[o + 1] * hn;
      bv.x = fast_tanh(Trow[o] * tn);
      bv.y = fast_tanh(Trow[o + 1] * tn);
      c = __builtin_amdgcn_wmma_f32_16x16x4_f32(false, av, false, bv,
                                                (short)0, c, false, false);
    }
    // diagonal D[i][i]: i<8 -> VGPR i, lane i ; i>=8 -> VGPR i-8, lane i+16
    int idx = lane & 7;
    float dv = c[0];
    dv = (idx == 1) ? c[1] : dv;
    dv = (idx == 2) ? c[2] : dv;
    dv = (idx == 3) ? c[3] : dv;
    dv = (idx == 4) ? c[4] : dv;
    dv = (idx == 5) ? c[5] : dv;
    dv = (idx == 6) ? c[6] : dv;
    dv = (idx == 7) ? c[7] : dv;
    if (active && eOut < E) A[(size_t)eOut * 4 + f] += dv;  // unique (e,f) writer
  }
}

// ---------------- epilogue ----------------
__global__ void k_acc_add(float* __restrict__ acc, const float* __restrict__ x, int n) {
  int i = blockIdx.x * blockDim.x + threadIdx.x;
  if (i < n) acc[i] += x[i];
}

__global__ void k_final(const float* __restrict__ acc, float* __restrict__ out, int n) {
  int i = blockIdx.x * blockDim.x + threadIdx.x;
  if (i < n) out[i] = acc[i] * (1.0f / 3.0f);
}

static inline size_t align16f(size_t x) { return (x + 15) & ~(size_t)15; }

extern "C" void kernel_launch(void* const* d_in, const int* in_sizes, int n_in,
                              void* d_out, int out_size, void* d_ws, size_t ws_size,
                              hipStream_t stream) {
  (void)n_in; (void)out_size; (void)ws_size;
  const float* u  = (const float*)d_in[0];
  const float* it = (const float*)d_in[1];
  const int* hl   = (const int*)d_in[2];
  const int* tl   = (const int*)d_in[3];

  const int userElems = in_sizes[0];
  const int itemElems = in_sizes[1];
  const int E = in_sizes[2];
  const int N = (userElems + itemElems) / 64;
  const size_t N64 = (size_t)N * 64;
  const size_t E4  = (size_t)E * 4;
  const size_t N4  = (size_t)N * 4;

  float* base = (float*)d_ws;
  size_t off = 0;
  float* ego   = base + off; off += align16f(N64);
  float* fact  = base + off; off += align16f(N64);
  float* acc   = base + off; off += align16f(N64);
  float* A     = base + off; off += align16f(E4);
  float* ew    = base + off; off += align16f(E4);
  float* dval  = base + off; off += align16f(N4);  // reused in place as dinv
  float* hninv = base + off; off += align16f(N4);
  float* tninv = base + off; off += align16f(N4);

#define GRID(n) ((unsigned)(((size_t)(n) + WG - 1) / WG))

  k_init<<<GRID(N64), WG, 0, stream>>>(u, it, ego, acc, userElems, (int)N64);
  k_fill1<<<GRID(E4), WG, 0, stream>>>(A, (int)E4);

  float* cur = ego;
  float* nxt = fact;
  const int nGroups = (E + 15) / 16;

  for (int layer = 0; layer < 2; ++layer) {
    // tail norms depend only on the layer-input embedding
    k_norminv<<<GRID(N4), WG, 0, stream>>>(cur, tninv, (int)N4);
    for (int itn = 0; itn < 2; ++itn) {
      hipMemsetAsync(dval, 0, N4 * sizeof(float), stream);
      k_softmax_dval<<<GRID(E), WG, 0, stream>>>(A, hl, ew, dval, E);
      k_dinv<<<GRID(N4), WG, 0, stream>>>(dval, (int)N4);
      k_edge_weight<<<GRID(E), WG, 0, stream>>>(ew, dval, hl, tl, E);
      hipMemsetAsync(nxt, 0, N64 * sizeof(float), stream);
      k_message<<<GRID((size_t)E * 16), WG, 0, stream>>>(ew, cur, hl, tl, nxt, E);
      const bool lastUpdateDead = (layer == 1 && itn == 1);  // A never read again
      if (!lastUpdateDead) {
        k_norminv<<<GRID(N4), WG, 0, stream>>>(nxt, hninv, (int)N4);
        k_updateA<<<GRID((size_t)nGroups * 32), WG, 0, stream>>>(
            nxt, cur, hninv, tninv, hl, tl, A, E, nGroups);
      }
    }
    k_acc_add<<<GRID(N64), WG, 0, stream>>>(acc, nxt, (int)N64);
    float* tmp = cur; cur = nxt; nxt = tmp;
  }

  k_final<<<GRID(N64), WG, 0, stream>>>(acc, (float*)d_out, (int)N64);
#undef GRID
}